// Decoder_3092376453486
// MI455X (gfx1250) — compile-verified
//
#include <hip/hip_runtime.h>
#include <hip/hip_bf16.h>
#include <math.h>

typedef __attribute__((ext_vector_type(2))) float v2f;
typedef __attribute__((ext_vector_type(8))) float v8f;

#define BATCH 64
#define SEQ   2048
#define HID   512
#define VOC   50257
#define KIN   1024   // I + H
#define NCHUNK 16
#define SCHUNK (SEQ / NCHUNK)   // 128 s-positions per block

// ---- CDNA5 fp32 WMMA: D(16x16) = A(16x4) * B(4x16) + C -------------------
// A layout: lane<16 -> M=lane, {a.x,a.y} = A[M][k+0..1]; lane>=16 -> M=lane-16,
//           {a.x,a.y} = A[M][k+2..3].  B layout mirrored over N.  C/D: vgpr i is
// row M=i (+8 for lanes>=16), lane%16 is column N.
__device__ __forceinline__ v8f wmma_f32(v2f a, v2f b, v8f c) {
  return __builtin_amdgcn_wmma_f32_16x16x4_f32(false, a, false, b, (short)0, c,
                                               false, false);
}

__device__ __forceinline__ float sigmoidf_(float x) {
  return 1.0f / (1.0f + __expf(-x));
}

// ---------------------------------------------------------------------------
// 1) Fused attention: scores + online softmax + context, one pass over enc.
//    grid (NCHUNK, BATCH), 256 threads. Partial per (b,chunk): [m, l, ctx[512]]
// ---------------------------------------------------------------------------
__global__ __launch_bounds__(256) void attn_flash_kernel(
    const float* __restrict__ h_top, const float* __restrict__ enc,
    float* __restrict__ partials) {
  __shared__ float sh_h[HID];
  __shared__ float sh_ctx[8][HID];
  __shared__ float sh_m[8];
  __shared__ float sh_l[8];

  const int b = blockIdx.y;
  const int chunk = blockIdx.x;
  for (int i = threadIdx.x; i < HID; i += 256) sh_h[i] = h_top[b * HID + i];
  __syncthreads();

  const int wave = threadIdx.x >> 5;
  const int lane = threadIdx.x & 31;
  const int h0 = 4 * lane;  // this lane owns h = 4*lane + 128*j, j=0..3

  const float4 hh0 = *(const float4*)&sh_h[h0];
  const float4 hh1 = *(const float4*)&sh_h[h0 + 128];
  const float4 hh2 = *(const float4*)&sh_h[h0 + 256];
  const float4 hh3 = *(const float4*)&sh_h[h0 + 384];

  float m = -INFINITY, l = 0.0f;
  float4 c0 = {0, 0, 0, 0}, c1 = {0, 0, 0, 0}, c2 = {0, 0, 0, 0},
         c3 = {0, 0, 0, 0};

  for (int t = 0; t < SCHUNK / 8; ++t) {
    const int s = chunk * SCHUNK + wave * (SCHUNK / 8) + t;
    const float* ep = enc + ((size_t)s * BATCH + b) * HID;
    const float4 e0 = *(const float4*)&ep[h0];
    const float4 e1 = *(const float4*)&ep[h0 + 128];
    const float4 e2 = *(const float4*)&ep[h0 + 256];
    const float4 e3 = *(const float4*)&ep[h0 + 384];

    float dot = e0.x * hh0.x + e0.y * hh0.y + e0.z * hh0.z + e0.w * hh0.w +
                e1.x * hh1.x + e1.y * hh1.y + e1.z * hh1.z + e1.w * hh1.w +
                e2.x * hh2.x + e2.y * hh2.y + e2.z * hh2.z + e2.w * hh2.w +
                e3.x * hh3.x + e3.y * hh3.y + e3.z * hh3.z + e3.w * hh3.w;
#pragma unroll
    for (int off = 16; off > 0; off >>= 1) dot += __shfl_xor(dot, off, 32);

    const float mn = fmaxf(m, dot);
    const float scale = __expf(m - mn);   // 0 on first iter (m = -inf)
    const float p = __expf(dot - mn);
    l = l * scale + p;
    c0.x = c0.x * scale + p * e0.x; c0.y = c0.y * scale + p * e0.y;
    c0.z = c0.z * scale + p * e0.z; c0.w = c0.w * scale + p * e0.w;
    c1.x = c1.x * scale + p * e1.x; c1.y = c1.y * scale + p * e1.y;
    c1.z = c1.z * scale + p * e1.z; c1.w = c1.w * scale + p * e1.w;
    c2.x = c2.x * scale + p * e2.x; c2.y = c2.y * scale + p * e2.y;
    c2.z = c2.z * scale + p * e2.z; c2.w = c2.w * scale + p * e2.w;
    c3.x = c3.x * scale + p * e3.x; c3.y = c3.y * scale + p * e3.y;
    c3.z = c3.z * scale + p * e3.z; c3.w = c3.w * scale + p * e3.w;
    m = mn;
  }

  if (lane == 0) { sh_m[wave] = m; sh_l[wave] = l; }
  *(float4*)&sh_ctx[wave][h0]       = c0;
  *(float4*)&sh_ctx[wave][h0 + 128] = c1;
  *(float4*)&sh_ctx[wave][h0 + 256] = c2;
  *(float4*)&sh_ctx[wave][h0 + 384] = c3;
  __syncthreads();

  float M = sh_m[0];
#pragma unroll
  for (int w = 1; w < 8; ++w) M = fmaxf(M, sh_m[w]);
  float wsc[8];
  float L = 0.0f;
#pragma unroll
  for (int w = 0; w < 8; ++w) {
    wsc[w] = __expf(sh_m[w] - M);
    L += sh_l[w] * wsc[w];
  }

  float* outp = partials + ((size_t)b * NCHUNK + chunk) * (HID + 2);
  for (int h = threadIdx.x; h < HID; h += 256) {
    float acc = 0.0f;
#pragma unroll
    for (int w = 0; w < 8; ++w) acc += sh_ctx[w][h] * wsc[w];
    outp[2 + h] = acc;
  }
  if (threadIdx.x == 0) { outp[0] = M; outp[1] = L; }
}

// ---------------------------------------------------------------------------
// 2) Combine chunk partials -> context; also gather embedding row.
//    grid BATCH, 256 threads.
// ---------------------------------------------------------------------------
__global__ __launch_bounds__(256) void attn_combine_kernel(
    const float* __restrict__ partials, const int* __restrict__ last_output,
    const float* __restrict__ embedding, float* __restrict__ rnn_in,
    float* __restrict__ feat) {
  const int b = blockIdx.x;
  const float* base = partials + (size_t)b * NCHUNK * (HID + 2);

  float M = -INFINITY;
#pragma unroll
  for (int c = 0; c < NCHUNK; ++c) M = fmaxf(M, base[(size_t)c * (HID + 2)]);
  float wsc[NCHUNK];
  float L = 0.0f;
#pragma unroll
  for (int c = 0; c < NCHUNK; ++c) {
    const float* pc = base + (size_t)c * (HID + 2);
    wsc[c] = __expf(pc[0] - M);
    L += pc[1] * wsc[c];
  }
  const float invL = 1.0f / L;
  const int tok = last_output[b];

  for (int h = threadIdx.x; h < HID; h += 256) {
    float acc = 0.0f;
#pragma unroll
    for (int c = 0; c < NCHUNK; ++c)
      acc += base[(size_t)c * (HID + 2) + 2 + h] * wsc[c];
    const float ctxv = acc * invL;
    rnn_in[(size_t)b * KIN + HID + h] = ctxv;
    feat[(size_t)b * KIN + HID + h] = ctxv;
    rnn_in[(size_t)b * KIN + h] = embedding[(size_t)tok * HID + h];
  }
}

// ---------------------------------------------------------------------------
// 3) Fused GRU cell via fp32 WMMA. grid HID/16, 128 threads (4 waves, one
//    16-batch M-tile each). 6 accumulator tiles: (r,z,n) x (input, hidden).
// ---------------------------------------------------------------------------
__global__ __launch_bounds__(128) void gru_kernel(
    const float* __restrict__ rnn_in, const float* __restrict__ h_top,
    const float* __restrict__ W_ih, const float* __restrict__ b_ih,
    const float* __restrict__ W_hh, const float* __restrict__ b_hh,
    float* __restrict__ feat, float* __restrict__ out_hidden) {
  const int wave = threadIdx.x >> 5;
  const int lane = threadIdx.x & 31;
  const int col = lane & 15;
  const int koff = (lane < 16) ? 0 : 2;
  const int mrow = wave * 16 + col;
  const int n = blockIdx.x * 16 + col;

  v8f air = {}, aiz = {}, ain = {}, ahr = {}, ahz = {}, ahn = {};

  const float* Ap = rnn_in + (size_t)mrow * KIN + koff;
  const float* Br = W_ih + (size_t)n * KIN + koff;
  const float* Bz = W_ih + (size_t)(n + HID) * KIN + koff;
  const float* Bn = W_ih + (size_t)(n + 2 * HID) * KIN + koff;
  for (int k = 0; k < KIN; k += 4) {
    const v2f a = *(const v2f*)(Ap + k);
    air = wmma_f32(a, *(const v2f*)(Br + k), air);
    aiz = wmma_f32(a, *(const v2f*)(Bz + k), aiz);
    ain = wmma_f32(a, *(const v2f*)(Bn + k), ain);
  }

  const float* Hp = h_top + (size_t)mrow * HID + koff;
  const float* Cr = W_hh + (size_t)n * HID + koff;
  const float* Cz = W_hh + (size_t)(n + HID) * HID + koff;
  const float* Cn = W_hh + (size_t)(n + 2 * HID) * HID + koff;
  for (int k = 0; k < HID; k += 4) {
    const v2f a = *(const v2f*)(Hp + k);
    ahr = wmma_f32(a, *(const v2f*)(Cr + k), ahr);
    ahz = wmma_f32(a, *(const v2f*)(Cz + k), ahz);
    ahn = wmma_f32(a, *(const v2f*)(Cn + k), ahn);
  }

  const float bir = b_ih[n], biz = b_ih[n + HID], bin_ = b_ih[n + 2 * HID];
  const float bhr = b_hh[n], bhz = b_hh[n + HID], bhn = b_hh[n + 2 * HID];
#pragma unroll
  for (int i = 0; i < 8; ++i) {
    const int row = wave * 16 + i + ((lane < 16) ? 0 : 8);
    const float r = sigmoidf_(air[i] + bir + ahr[i] + bhr);
    const float z = sigmoidf_(aiz[i] + biz + ahz[i] + bhz);
    const float nn = tanhf(ain[i] + bin_ + r * (ahn[i] + bhn));
    const float hp = h_top[(size_t)row * HID + n];
    const float hv = (1.0f - z) * nn + z * hp;
    feat[(size_t)row * KIN + n] = hv;
    out_hidden[(size_t)row * HID + n] = hv;
  }
}

// ---------------------------------------------------------------------------
// 4) logits = feat @ W_out^T + b_out via fp32 WMMA, written into d_out logp
//    region. grid ceil(VOC/16), 128 threads. Each W_out element read once.
// ---------------------------------------------------------------------------
__global__ __launch_bounds__(128) void logits_kernel(
    const float* __restrict__ feat, const float* __restrict__ W_out,
    const float* __restrict__ b_out, float* __restrict__ logits) {
  const int wave = threadIdx.x >> 5;
  const int lane = threadIdx.x & 31;
  const int col = lane & 15;
  const int koff = (lane < 16) ? 0 : 2;
  const int mrow = wave * 16 + col;
  const int v = blockIdx.x * 16 + col;
  const int vc = (v < VOC) ? v : (VOC - 1);

  v8f acc0 = {}, acc1 = {};
  const float* Ap = feat + (size_t)mrow * KIN + koff;
  const float* Bp = W_out + (size_t)vc * KIN + koff;
  for (int k = 0; k < KIN; k += 8) {  // two independent WMMA chains for ILP
    acc0 = wmma_f32(*(const v2f*)(Ap + k), *(const v2f*)(Bp + k), acc0);
    acc1 = wmma_f32(*(const v2f*)(Ap + k + 4), *(const v2f*)(Bp + k + 4), acc1);
  }
  const v8f acc = acc0 + acc1;

  if (v < VOC) {
    const float bo = b_out[v];
#pragma unroll
    for (int i = 0; i < 8; ++i) {
      const int row = wave * 16 + i + ((lane < 16) ? 0 : 8);
      logits[(size_t)row * VOC + v] = acc[i] + bo;
    }
  }
}

// ---------------------------------------------------------------------------
// 5) Row-wise logsumexp over VOC. grid BATCH, 256 threads.
// ---------------------------------------------------------------------------
__global__ __launch_bounds__(256) void lse_kernel(
    const float* __restrict__ logits, float* __restrict__ lse) {
  __shared__ float red[256];
  const int b = blockIdx.x;
  const float* row = logits + (size_t)b * VOC;

  float m = -INFINITY;
  for (int v = threadIdx.x; v < VOC; v += 256) m = fmaxf(m, row[v]);
  red[threadIdx.x] = m;
  __syncthreads();
  for (int off = 128; off > 0; off >>= 1) {
    if (threadIdx.x < off)
      red[threadIdx.x] = fmaxf(red[threadIdx.x], red[threadIdx.x + off]);
    __syncthreads();
  }
  const float M = red[0];
  __syncthreads();

  float s = 0.0f;
  for (int v = threadIdx.x; v < VOC; v += 256) s += __expf(row[v] - M);
  red[threadIdx.x] = s;
  __syncthreads();
  for (int off = 128; off > 0; off >>= 1) {
    if (threadIdx.x < off) red[threadIdx.x] += red[threadIdx.x + off];
    __syncthreads();
  }
  if (threadIdx.x == 0) lse[b] = M + __logf(red[0]);
}

// 6) In-place logp = logits - lse[b]. grid (ceil(VOC/256), BATCH).
__global__ __launch_bounds__(256) void logp_kernel(
    float* __restrict__ logp, const float* __restrict__ lse) {
  const int b = blockIdx.y;
  const int v = blockIdx.x * 256 + threadIdx.x;
  if (v < VOC) logp[(size_t)b * VOC + v] -= lse[b];
}

// ---------------------------------------------------------------------------
extern "C" void kernel_launch(void* const* d_in, const int* in_sizes, int n_in,
                              void* d_out, int out_size, void* d_ws,
                              size_t ws_size, hipStream_t stream) {
  (void)in_sizes; (void)n_in; (void)out_size; (void)ws_size;
  const int* last_output   = (const int*)d_in[0];
  const float* last_hidden = (const float*)d_in[1];  // (1,B,H) -> B*H
  const float* enc         = (const float*)d_in[2];
  const float* embedding   = (const float*)d_in[3];
  const float* W_ih        = (const float*)d_in[4];
  const float* b_ih        = (const float*)d_in[5];
  const float* W_hh        = (const float*)d_in[6];
  const float* b_hh        = (const float*)d_in[7];
  const float* W_out       = (const float*)d_in[8];
  const float* b_out       = (const float*)d_in[9];

  float* logp = (float*)d_out;                         // B*VOC
  float* out_hidden = logp + (size_t)BATCH * VOC;      // B*HID

  float* ws = (float*)d_ws;
  float* partials = ws;                                       // B*NCHUNK*(HID+2)
  float* rnn_in = partials + (size_t)BATCH * NCHUNK * (HID + 2);  // B*KIN
  float* feat = rnn_in + (size_t)BATCH * KIN;                 // B*KIN
  float* lse = feat + (size_t)BATCH * KIN;                    // B

  attn_flash_kernel<<<dim3(NCHUNK, BATCH), 256, 0, stream>>>(last_hidden, enc,
                                                             partials);
  attn_combine_kernel<<<BATCH, 256, 0, stream>>>(partials, last_output,
                                                 embedding, rnn_in, feat);
  gru_kernel<<<HID / 16, 128, 0, stream>>>(rnn_in, last_hidden, W_ih, b_ih,
                                           W_hh, b_hh, feat, out_hidden);
  logits_kernel<<<(VOC + 15) / 16, 128, 0, stream>>>(feat, W_out, b_out, logp);
  lse_kernel<<<BATCH, 256, 0, stream>>>(logp, lse);
  logp_kernel<<<dim3((VOC + 255) / 256, BATCH), 256, 0, stream>>>(logp, lse);
}